// DSSM_25993142075727
// MI455X (gfx1250) — compile-verified
//
#include <hip/hip_runtime.h>
#include <hip/hip_bf16.h>
#include <math.h>

// ---------------------------------------------------------------------------
// DSSM fused pipeline for gfx1250 (MI455X). fp32 WMMA (V_WMMA_F32_16X16X4_F32)
// for all dense projections; per-channel sequential selective scan with
// global_prefetch for the serial dependency chain.
// ---------------------------------------------------------------------------

typedef float v2f __attribute__((ext_vector_type(2)));
typedef float v8f __attribute__((ext_vector_type(8)));

#define NPIX   2304      // 48*48
#define DM     96        // D_MODEL
#define DI     192       // D_INNER
#define LSEQ   4608      // 2*NPIX
#define NST    4         // D_STATE
#define RDT    6         // DT_RANK

__device__ __forceinline__ float sigmoidf_(float x) { return 1.0f / (1.0f + expf(-x)); }
__device__ __forceinline__ v2f   ldv2(const float* p) { return *(const v2f*)p; }

#define WMMA_F32(ACC, A, B) \
    __builtin_amdgcn_wmma_f32_16x16x4_f32(false, (A), false, (B), (short)0, (ACC), false, false)

// ---------------------------------------------------------------------------
// Kernel 1: input projections via WMMA f32 16x16x4, K=96.
//  mode 0: xvz[2304x384] = x_vi @ W_vi^T
//  mode 1: xiz[2304x384] = x_ir @ W_ir^T
//  mode 2: xsub[2304x192] = (x_vi - x_ir) @ W_sub^T
// All 24 K-step fragments are batch-loaded, then the WMMA chain runs
// back-to-back (one wait instead of 24).
// ---------------------------------------------------------------------------
__global__ __launch_bounds__(32) void gemm_in_kernel(
    const float* __restrict__ xvi, const float* __restrict__ xir,
    const float* __restrict__ Wvi, const float* __restrict__ Wir,
    const float* __restrict__ Wsub,
    float* __restrict__ xvz, float* __restrict__ xiz, float* __restrict__ xsub)
{
    int lane = threadIdx.x;
    int mt = blockIdx.x, nt = blockIdx.y, mode = blockIdx.z;
    if (mode == 2 && nt >= 12) return;          // wave-uniform early exit
    int m0 = mt * 16, n0 = nt * 16;
    int mr = lane & 15;                          // row of A / row of W
    int kh = (lane >> 4) << 1;                   // K offset {0,2}

    const float* W = (mode == 0) ? Wvi : (mode == 1) ? Wir : Wsub;
    const float* Arow0 = xvi + (size_t)(m0 + mr) * DM + kh;
    const float* Arow1 = xir + (size_t)(m0 + mr) * DM + kh;
    const float* Brow  = W   + (size_t)(n0 + mr) * DM + kh;

    v2f af[24], bf[24];
    if (mode == 0) {
        #pragma unroll
        for (int t = 0; t < 24; ++t) { af[t] = ldv2(Arow0 + t*4); bf[t] = ldv2(Brow + t*4); }
    } else if (mode == 1) {
        #pragma unroll
        for (int t = 0; t < 24; ++t) { af[t] = ldv2(Arow1 + t*4); bf[t] = ldv2(Brow + t*4); }
    } else {
        #pragma unroll
        for (int t = 0; t < 24; ++t) { af[t] = ldv2(Arow0 + t*4) - ldv2(Arow1 + t*4); bf[t] = ldv2(Brow + t*4); }
    }

    v8f acc = {0.f,0.f,0.f,0.f,0.f,0.f,0.f,0.f};
    #pragma unroll
    for (int t = 0; t < 24; ++t) acc = WMMA_F32(acc, af[t], bf[t]);

    float* C   = (mode == 0) ? xvz : (mode == 1) ? xiz : xsub;
    int    ldc = (mode == 2) ? DI : 2*DI;
    int chi = (lane >> 4) * 8;
    #pragma unroll
    for (int i = 0; i < 8; ++i)
        C[(size_t)(m0 + chi + i) * ldc + n0 + mr] = acc[i];
}

// ---------------------------------------------------------------------------
// Kernel 2: channel attention scale s[c] for vi (g=0) and ir (g=1).
// ---------------------------------------------------------------------------
__global__ void ca_kernel(const float* __restrict__ xvz, const float* __restrict__ xiz,
                          const float* __restrict__ f1vi, const float* __restrict__ f2vi,
                          const float* __restrict__ f1ir, const float* __restrict__ f2ir,
                          float* __restrict__ ca_s)
{
    __shared__ float avg[DI], mx[DI], ha[12], hm[12];
    int g = blockIdx.x, c = threadIdx.x;
    const float* Z  = (g == 0) ? xvz  : xiz;
    const float* f1 = (g == 0) ? f1vi : f1ir;
    const float* f2 = (g == 0) ? f2vi : f2ir;

    float s = 0.f, m = -INFINITY;
    for (int p = 0; p < NPIX; ++p) {
        float v  = Z[(size_t)p*(2*DI) + DI + c];
        float sv = v * sigmoidf_(v);
        s += sv; m = fmaxf(m, sv);
    }
    avg[c] = s * (1.0f / (float)NPIX);
    mx[c]  = m;
    __syncthreads();
    if (c < 12) {
        float sa = 0.f, sm = 0.f;
        for (int j = 0; j < DI; ++j) {
            float w = f1[c*DI + j];
            sa += avg[j] * w; sm += mx[j] * w;
        }
        ha[c] = fmaxf(sa, 0.f); hm[c] = fmaxf(sm, 0.f);
    }
    __syncthreads();
    float oa = 0.f, om = 0.f;
    #pragma unroll
    for (int j = 0; j < 12; ++j) {
        float w = f2[c*12 + j];
        oa += ha[j] * w; om += hm[j] * w;
    }
    ca_s[g*DI + c] = sigmoidf_(oa + om);
}

// ---------------------------------------------------------------------------
// Kernel 3: depthwise 3x3 conv (SAME) + bias + SiLU.
// ---------------------------------------------------------------------------
__global__ void dwconv_kernel(const float* __restrict__ xvz, const float* __restrict__ xiz,
                              const float* __restrict__ xsub,
                              const float* __restrict__ wv, const float* __restrict__ bv,
                              const float* __restrict__ wi, const float* __restrict__ bi,
                              const float* __restrict__ ws_, const float* __restrict__ bs,
                              float* __restrict__ xv_c, float* __restrict__ xi_c,
                              float* __restrict__ xs_c)
{
    int idx = blockIdx.x * 256 + threadIdx.x;
    if (idx >= NPIX * DI) return;
    int mode = blockIdx.y;
    int c = idx % DI, p = idx / DI;
    int h = p / 48, w = p % 48;

    const float* In; int ld; const float* Wt; const float* Bi; float* Out;
    if (mode == 0)      { In = xvz;  ld = 2*DI; Wt = wv;  Bi = bv; Out = xv_c; }
    else if (mode == 1) { In = xiz;  ld = 2*DI; Wt = wi;  Bi = bi; Out = xi_c; }
    else                { In = xsub; ld = DI;   Wt = ws_; Bi = bs; Out = xs_c; }

    float acc = Bi[c];
    #pragma unroll
    for (int dy = 0; dy < 3; ++dy) {
        int hh = h + dy - 1; if (hh < 0 || hh >= 48) continue;
        #pragma unroll
        for (int dx = 0; dx < 3; ++dx) {
            int ww = w + dx - 1; if (ww < 0 || ww >= 48) continue;
            acc += In[(size_t)(hh*48 + ww)*ld + c] * Wt[c*9 + dy*3 + dx];
        }
    }
    Out[(size_t)p*DI + c] = acc * sigmoidf_(acc);
}

// ---------------------------------------------------------------------------
// Kernel 4: x_dbl[k][l][16] = s_k[l][:] @ x_proj_weight[k]^T (c=14, pad 16).
// K=192, pipelined in chunks of 16 WMMA steps (64 fragment VGPRs).
// ---------------------------------------------------------------------------
__global__ __launch_bounds__(32) void gemm_xdbl_kernel(
    const float* __restrict__ xs_c, const float* __restrict__ xv_c,
    const float* __restrict__ xi_c, const float* __restrict__ xpw,
    float* __restrict__ xdbl)
{
    int lane = threadIdx.x;
    int mt = blockIdx.x, k = blockIdx.y;
    int m0 = mt * 16;
    int mr = lane & 15, kh = (lane >> 4) << 1;

    int l = m0 + mr;                 // tiles never straddle l=2304 (16 | 2304)
    const float* Arow; int lr;
    if (l < NPIX) { Arow = xs_c; lr = l; }
    else          { Arow = (k & 1) ? xi_c : xv_c; lr = l - NPIX; }
    const float* Ap = Arow + (size_t)lr * DI + kh;
    const float* Wp = xpw + (size_t)k * 14 * DI + (size_t)mr * DI + kh;
    bool bvalid = (mr < 14);

    v8f acc = {0.f,0.f,0.f,0.f,0.f,0.f,0.f,0.f};
    #pragma unroll
    for (int kb = 0; kb < DI; kb += 64) {
        v2f af[16], bf[16];
        #pragma unroll
        for (int t = 0; t < 16; ++t) {
            af[t] = ldv2(Ap + kb + t*4);
            if (bvalid) bf[t] = ldv2(Wp + kb + t*4);
            else        { bf[t].x = 0.f; bf[t].y = 0.f; }
        }
        #pragma unroll
        for (int t = 0; t < 16; ++t) acc = WMMA_F32(acc, af[t], bf[t]);
    }
    float* Co = xdbl + (size_t)k * LSEQ * 16;
    int chi = (lane >> 4) * 8;
    #pragma unroll
    for (int i = 0; i < 8; ++i)
        Co[(size_t)(m0 + chi + i) * 16 + mr] = acc[i];
}

// ---------------------------------------------------------------------------
// Kernel 5: selective scan. One lane per (k,d): 768 lanes = 24 wave32s.
// k<2 forward over full L; k>=2 backward, stopping at l=NPIX. Next-step rows
// are prefetched (global_prefetch) to hide latency in the serial chain.
// ---------------------------------------------------------------------------
__global__ __launch_bounds__(32) void scan_kernel(
    const float* __restrict__ xdbl,
    const float* __restrict__ xs_c, const float* __restrict__ xv_c,
    const float* __restrict__ xi_c,
    const float* __restrict__ dtw_all, const float* __restrict__ dtb_all,
    const float* __restrict__ A_logs, const float* __restrict__ Ds,
    float* __restrict__ yv0, float* __restrict__ yv2,
    float* __restrict__ yi1, float* __restrict__ yi3)
{
    int gid = blockIdx.x * 32 + threadIdx.x;   // 0..767 ; k uniform per wave
    int k = gid / DI, d = gid % DI;

    float dtw[RDT];
    #pragma unroll
    for (int r = 0; r < RDT; ++r) dtw[r] = dtw_all[(size_t)gid*RDT + r];
    float dtb = dtb_all[gid];
    float An[NST], hn[NST] = {0.f, 0.f, 0.f, 0.f};
    #pragma unroll
    for (int n = 0; n < NST; ++n) An[n] = -expf(A_logs[(size_t)gid*NST + n]);
    float Dd = Ds[gid];

    const float* sx = (k & 1) ? xi_c : xv_c;
    const float* Xd = xdbl + (size_t)k * LSEQ * 16;
    float* Out = (k == 0) ? yv0 : (k == 1) ? yi1 : (k == 2) ? yv2 : yi3;
    bool fwd = (k < 2);
    int nsteps = fwd ? LSEQ : NPIX;            // backward only needs l>=NPIX
    int dir = fwd ? 1 : -1;

    for (int step = 0; step < nsteps; ++step) {
        int l = fwd ? step : (LSEQ - 1 - step);
        const float* xr = Xd + (size_t)l * 16;
        // prefetch next step's data into cache while this step computes
        int ln = l + dir;
        if (ln >= 0 && ln < LSEQ) {
            __builtin_prefetch(Xd + (size_t)ln * 16, 0, 3);
            const float* un = (ln < NPIX) ? (xs_c + (size_t)ln*DI + d)
                                          : (sx + (size_t)(ln - NPIX)*DI + d);
            __builtin_prefetch(un, 0, 3);
        }
        float u = (l < NPIX) ? xs_c[(size_t)l*DI + d]
                             : sx[(size_t)(l - NPIX)*DI + d];
        float draw = dtb;
        #pragma unroll
        for (int r = 0; r < RDT; ++r) draw += xr[r] * dtw[r];
        float delta = (draw > 20.f) ? draw : log1pf(expf(draw));
        float du = delta * u;
        float y = Dd * u;
        #pragma unroll
        for (int n = 0; n < NST; ++n) {
            float dA = expf(delta * An[n]);
            hn[n] = dA * hn[n] + du * xr[6 + n];   // B at c=6..9
            y += hn[n] * xr[10 + n];               // C at c=10..13
        }
        if (l >= NPIX)
            Out[(size_t)(l - NPIX)*DI + d] = y;
    }
}

// ---------------------------------------------------------------------------
// Kernel 6: per-pixel LayerNorm(y_vi), LayerNorm(y_ir), gate with
// (1+ca_s)*silu(z), write fused[2304x192].
// ---------------------------------------------------------------------------
__global__ void ln_fuse_kernel(
    const float* __restrict__ yv0, const float* __restrict__ yv2,
    const float* __restrict__ yi1, const float* __restrict__ yi3,
    const float* __restrict__ xvz, const float* __restrict__ xiz,
    const float* __restrict__ ca_s,
    const float* __restrict__ gvi, const float* __restrict__ bvi,
    const float* __restrict__ gir, const float* __restrict__ bir,
    float* __restrict__ fused)
{
    __shared__ float red[256];
    int p = blockIdx.x, t = threadIdx.x;
    float yv = yv0[(size_t)p*DI + t] + yv2[(size_t)p*DI + t];
    float yi = yi1[(size_t)p*DI + t] + yi3[(size_t)p*DI + t];

    auto reduce = [&](float v) -> float {
        red[t] = v;
        if (t < 64) red[DI + t] = 0.f;
        __syncthreads();
        for (int s2 = 128; s2 > 0; s2 >>= 1) {
            if (t < s2) red[t] += red[t + s2];
            __syncthreads();
        }
        float r = red[0];
        __syncthreads();
        return r;
    };

    float mv = reduce(yv) * (1.0f / (float)DI);
    float vv = reduce((yv - mv)*(yv - mv)) * (1.0f / (float)DI);
    float mi = reduce(yi) * (1.0f / (float)DI);
    float vi = reduce((yi - mi)*(yi - mi)) * (1.0f / (float)DI);

    float yvn = (yv - mv) * rsqrtf(vv + 1e-5f) * gvi[t] + bvi[t];
    float yin = (yi - mi) * rsqrtf(vi + 1e-5f) * gir[t] + bir[t];

    float zv = xvz[(size_t)p*(2*DI) + DI + t]; zv = zv * sigmoidf_(zv);
    float zi = xiz[(size_t)p*(2*DI) + DI + t]; zi = zi * sigmoidf_(zi);
    float zvf = zv + ca_s[t]       * zv;
    float zif = zi + ca_s[DI + t]  * zi;

    fused[(size_t)p*DI + t] = yvn * zvf + yin * zif;
}

// ---------------------------------------------------------------------------
// Kernel 7: out[2304x96] = fused @ W_out^T via WMMA f32 16x16x4 (K=192)
// ---------------------------------------------------------------------------
__global__ __launch_bounds__(32) void gemm_out_kernel(
    const float* __restrict__ fused, const float* __restrict__ Wout,
    float* __restrict__ out)
{
    int lane = threadIdx.x;
    int m0 = blockIdx.x * 16, n0 = blockIdx.y * 16;
    int mr = lane & 15, kh = (lane >> 4) << 1;
    const float* Ap = fused + (size_t)(m0 + mr)*DI + kh;
    const float* Bp = Wout  + (size_t)(n0 + mr)*DI + kh;

    v8f acc = {0.f,0.f,0.f,0.f,0.f,0.f,0.f,0.f};
    #pragma unroll
    for (int kb = 0; kb < DI; kb += 64) {
        v2f af[16], bf[16];
        #pragma unroll
        for (int t = 0; t < 16; ++t) {
            af[t] = ldv2(Ap + kb + t*4);
            bf[t] = ldv2(Bp + kb + t*4);
        }
        #pragma unroll
        for (int t = 0; t < 16; ++t) acc = WMMA_F32(acc, af[t], bf[t]);
    }
    int chi = (lane >> 4) * 8;
    #pragma unroll
    for (int i = 0; i < 8; ++i)
        out[(size_t)(m0 + chi + i)*DM + n0 + mr] = acc[i];
}

// ---------------------------------------------------------------------------
extern "C" void kernel_launch(void* const* d_in, const int* in_sizes, int n_in,
                              void* d_out, int out_size, void* d_ws, size_t ws_size,
                              hipStream_t stream) {
    const float* x_vi     = (const float*)d_in[0];
    const float* x_ir     = (const float*)d_in[1];
    const float* W_vi     = (const float*)d_in[2];
    const float* W_ir     = (const float*)d_in[3];
    const float* W_sub    = (const float*)d_in[4];
    const float* conv_w_vi  = (const float*)d_in[5];
    const float* conv_b_vi  = (const float*)d_in[6];
    const float* conv_w_ir  = (const float*)d_in[7];
    const float* conv_b_ir  = (const float*)d_in[8];
    const float* conv_w_sub = (const float*)d_in[9];
    const float* conv_b_sub = (const float*)d_in[10];
    const float* x_proj_weight  = (const float*)d_in[11];
    const float* dt_projs_weight = (const float*)d_in[12];
    const float* dt_projs_bias   = (const float*)d_in[13];
    const float* A_logs   = (const float*)d_in[14];
    const float* Ds       = (const float*)d_in[15];
    const float* ln_vi_g  = (const float*)d_in[16];
    const float* ln_vi_b  = (const float*)d_in[17];
    const float* ln_ir_g  = (const float*)d_in[18];
    const float* ln_ir_b  = (const float*)d_in[19];
    const float* W_out    = (const float*)d_in[20];
    const float* ca_vi_f1 = (const float*)d_in[21];
    const float* ca_vi_f2 = (const float*)d_in[22];
    const float* ca_ir_f1 = (const float*)d_in[23];
    const float* ca_ir_f2 = (const float*)d_in[24];
    float* out = (float*)d_out;

    // workspace layout (floats)
    float* w    = (float*)d_ws;
    float* xvz  = w;                       // 2304*384
    float* xiz  = xvz  + (size_t)NPIX*2*DI;
    float* xsub = xiz  + (size_t)NPIX*2*DI;    // 2304*192
    float* xv_c = xsub + (size_t)NPIX*DI;
    float* xi_c = xv_c + (size_t)NPIX*DI;
    float* xs_c = xi_c + (size_t)NPIX*DI;
    float* xdbl = xs_c + (size_t)NPIX*DI;      // 4*4608*16
    float* yv0  = xdbl + (size_t)4*LSEQ*16;
    float* yv2  = yv0  + (size_t)NPIX*DI;
    float* yi1  = yv2  + (size_t)NPIX*DI;
    float* yi3  = yi1  + (size_t)NPIX*DI;
    float* ca_s = yi3  + (size_t)NPIX*DI;      // 384
    float* fused = ca_s + 2*DI;                // 2304*192

    gemm_in_kernel<<<dim3(NPIX/16, 24, 3), 32, 0, stream>>>(
        x_vi, x_ir, W_vi, W_ir, W_sub, xvz, xiz, xsub);

    ca_kernel<<<2, DI, 0, stream>>>(
        xvz, xiz, ca_vi_f1, ca_vi_f2, ca_ir_f1, ca_ir_f2, ca_s);

    dwconv_kernel<<<dim3((NPIX*DI + 255)/256, 3), 256, 0, stream>>>(
        xvz, xiz, xsub,
        conv_w_vi, conv_b_vi, conv_w_ir, conv_b_ir, conv_w_sub, conv_b_sub,
        xv_c, xi_c, xs_c);

    gemm_xdbl_kernel<<<dim3(LSEQ/16, 4), 32, 0, stream>>>(
        xs_c, xv_c, xi_c, x_proj_weight, xdbl);

    scan_kernel<<<24, 32, 0, stream>>>(
        xdbl, xs_c, xv_c, xi_c,
        dt_projs_weight, dt_projs_bias, A_logs, Ds,
        yv0, yv2, yi1, yi3);

    ln_fuse_kernel<<<NPIX, DI, 0, stream>>>(
        yv0, yv2, yi1, yi3, xvz, xiz, ca_s,
        ln_vi_g, ln_vi_b, ln_ir_g, ln_ir_b, fused);

    gemm_out_kernel<<<dim3(NPIX/16, DM/16), 32, 0, stream>>>(
        fused, W_out, out);

    (void)in_sizes; (void)n_in; (void)out_size; (void)ws_size;
}